// SimpleMambaBlock_68281390072180
// MI455X (gfx1250) — compile-verified
//
#include <hip/hip_runtime.h>
#include <cstdint>
#include <cstddef>

// ---------------- problem constants (from reference) ----------------
#define B_SZ     2
#define L_SEQ    2048
#define DMODEL   1024
#define INNER    2048          // EXPAND * D_MODEL
#define DTRANK   64            // D_MODEL / 16
#define NSTATE   16
#define MROWS    (B_SZ * L_SEQ)        // 4096 tokens
#define XDBL_N   (DTRANK + 2 * NSTATE) // 96

// LDS staging geometry for the big GEMMs
#define BM 64                  // block tile M (2 waves of 32)
#define BN 128                 // block tile N (4 waves of 32)
#define KC 32                  // K panel per stage
#define LDSP 36                // padded row stride (floats): 144B = 16B-aligned,
                               // and stride-36 word addressing -> conflict-free
                               // v2f fragment reads (9n mod 16 is a permutation)

typedef float v2f __attribute__((ext_vector_type(2)));
typedef float v4f __attribute__((ext_vector_type(4)));
typedef float v8f __attribute__((ext_vector_type(8)));

__device__ __forceinline__ float softplus_f(float x) {
    return (x > 20.0f) ? x : log1pf(expf(x));
}
__device__ __forceinline__ float silu_f(float x) {
    return x / (1.0f + expf(-x));
}

__device__ __forceinline__ v8f wmma4(v2f a, v2f b, v8f c) {
    return __builtin_amdgcn_wmma_f32_16x16x4_f32(false, a, false, b,
                                                 (short)0, c, false, false);
}

// =====================================================================
// LDS-staged, double-buffered f32 WMMA GEMM:
//   C[M,N] = A[M,K] (row-major, leading dim lda) * W[N,K]^T (row-major, ld K)
// Block: 256 threads = 8 waves (2 M x 4 N), tile 64x128, K panel 32.
// mode 1: C = softplus(softplus(C + bias[n]))  (dt path, softplus twice as
// in the reference). grid = (N/128, M/64); K % 32 == 0.
// =====================================================================
__global__ __launch_bounds__(256) void wmma_gemm_staged_kernel(
    const float* __restrict__ A, const float* __restrict__ W,
    const float* __restrict__ bias, float* __restrict__ C,
    int N, int K, int lda, int mode)
{
    __shared__ float As[2][BM * LDSP];   // 2 x 9.2 KB
    __shared__ float Ws[2][BN * LDSP];   // 2 x 18.4 KB   (55.3 KB total)

    const int tid  = threadIdx.x;
    const int lane = tid & 31;
    const int wave = tid >> 5;
    const int wm = wave & 1;        // 0..1  (M direction)
    const int wn = wave >> 1;       // 0..3  (N direction)

    const int mblk = blockIdx.y * BM;
    const int nblk = blockIdx.x * BN;

    // ---- cooperative panel-load helpers (flat index -> row/col) ----
    // A panel: 64 rows x 32 floats = 512 v4f ; 2 per thread
    const int fa0 = tid, fa1 = tid + 256;
    const int ar0 = fa0 >> 3, ac0 = (fa0 & 7) << 2;
    const int ar1 = fa1 >> 3, ac1 = (fa1 & 7) << 2;
    // W panel: 128 rows x 32 floats = 1024 v4f ; 4 per thread
    const int wr0 = (tid + 0)   >> 3, wc0 = ((tid + 0)   & 7) << 2;
    const int wr1 = (tid + 256) >> 3, wc1 = ((tid + 256) & 7) << 2;
    const int wr2 = (tid + 512) >> 3, wc2 = ((tid + 512) & 7) << 2;
    const int wr3 = (tid + 768) >> 3, wc3 = ((tid + 768) & 7) << 2;

    const float* Abase = A + (size_t)mblk * lda;
    const float* Wbase = W + (size_t)nblk * K;

    v4f aReg0, aReg1, wReg0, wReg1, wReg2, wReg3;

    // prefetch chunk 0 -> regs
    {
        const int k0 = 0;
        aReg0 = *(const v4f*)(Abase + (size_t)ar0 * lda + k0 + ac0);
        aReg1 = *(const v4f*)(Abase + (size_t)ar1 * lda + k0 + ac1);
        wReg0 = *(const v4f*)(Wbase + (size_t)wr0 * K + k0 + wc0);
        wReg1 = *(const v4f*)(Wbase + (size_t)wr1 * K + k0 + wc1);
        wReg2 = *(const v4f*)(Wbase + (size_t)wr2 * K + k0 + wc2);
        wReg3 = *(const v4f*)(Wbase + (size_t)wr3 * K + k0 + wc3);
    }
    // regs -> LDS buf 0 (16B-aligned: LDSP*4 = 144, col*4 = 16c)
    *(v4f*)(&As[0][ar0 * LDSP + ac0]) = aReg0;
    *(v4f*)(&As[0][ar1 * LDSP + ac1]) = aReg1;
    *(v4f*)(&Ws[0][wr0 * LDSP + wc0]) = wReg0;
    *(v4f*)(&Ws[0][wr1 * LDSP + wc1]) = wReg1;
    *(v4f*)(&Ws[0][wr2 * LDSP + wc2]) = wReg2;
    *(v4f*)(&Ws[0][wr3 * LDSP + wc3]) = wReg3;
    __syncthreads();

    v8f acc00 = {}; v8f acc01 = {}; v8f acc10 = {}; v8f acc11 = {};

    // fragment read addresses within LDS panels (f32 16x16x4 layout:
    // lanes 0-15 -> K={k,k+1}, lanes 16-31 -> K={k+2,k+3})
    const int l15  = lane & 15;
    const int koff = (lane >> 4) << 1;
    const int aoff = (wm * 32 + l15) * LDSP + koff;
    const int woff = (wn * 32 + l15) * LDSP + koff;

    const int nk = K / KC;
    for (int kc = 0; kc < nk; ++kc) {
        const int buf = kc & 1;
        const bool has_next = (kc + 1) < nk;
        if (has_next) {
            const int k0 = (kc + 1) * KC;
            aReg0 = *(const v4f*)(Abase + (size_t)ar0 * lda + k0 + ac0);
            aReg1 = *(const v4f*)(Abase + (size_t)ar1 * lda + k0 + ac1);
            wReg0 = *(const v4f*)(Wbase + (size_t)wr0 * K + k0 + wc0);
            wReg1 = *(const v4f*)(Wbase + (size_t)wr1 * K + k0 + wc1);
            wReg2 = *(const v4f*)(Wbase + (size_t)wr2 * K + k0 + wc2);
            wReg3 = *(const v4f*)(Wbase + (size_t)wr3 * K + k0 + wc3);
        }

        const float* Ap = &As[buf][aoff];
        const float* Wp = &Ws[buf][woff];
#pragma unroll
        for (int kk = 0; kk < KC; kk += 4) {
            v2f a0 = *(const v2f*)(Ap + kk);
            v2f a1 = *(const v2f*)(Ap + 16 * LDSP + kk);
            v2f b0 = *(const v2f*)(Wp + kk);
            v2f b1 = *(const v2f*)(Wp + 16 * LDSP + kk);
            acc00 = wmma4(a0, b0, acc00);
            acc01 = wmma4(a0, b1, acc01);
            acc10 = wmma4(a1, b0, acc10);
            acc11 = wmma4(a1, b1, acc11);
        }

        if (has_next) {
            const int nb = buf ^ 1;
            *(v4f*)(&As[nb][ar0 * LDSP + ac0]) = aReg0;
            *(v4f*)(&As[nb][ar1 * LDSP + ac1]) = aReg1;
            *(v4f*)(&Ws[nb][wr0 * LDSP + wc0]) = wReg0;
            *(v4f*)(&Ws[nb][wr1 * LDSP + wc1]) = wReg1;
            *(v4f*)(&Ws[nb][wr2 * LDSP + wc2]) = wReg2;
            *(v4f*)(&Ws[nb][wr3 * LDSP + wc3]) = wReg3;
        }
        __syncthreads();
    }

    // ---- store (C/D layout: VGPR v -> M = base + (lane>=16?8:0) + v) ----
    const int m0 = mblk + wm * 32;
    const int n0 = nblk + wn * 32;
    const int mb  = m0 + ((lane >> 4) << 3);
    const int nc0 = n0 + l15;
    const int nc1 = nc0 + 16;
    float bias0 = 0.f, bias1 = 0.f;
    if (mode == 1) { bias0 = bias[nc0]; bias1 = bias[nc1]; }

#pragma unroll
    for (int v = 0; v < 8; ++v) {
        float c00 = acc00[v], c01 = acc01[v], c10 = acc10[v], c11 = acc11[v];
        if (mode == 1) {
            c00 = softplus_f(softplus_f(c00 + bias0));
            c01 = softplus_f(softplus_f(c01 + bias1));
            c10 = softplus_f(softplus_f(c10 + bias0));
            c11 = softplus_f(softplus_f(c11 + bias1));
        }
        C[(size_t)(mb + v)      * N + nc0] = c00;
        C[(size_t)(mb + v)      * N + nc1] = c01;
        C[(size_t)(mb + 16 + v) * N + nc0] = c10;
        C[(size_t)(mb + 16 + v) * N + nc1] = c11;
    }
}

// =====================================================================
// Direct-from-global f32 WMMA GEMM (used for the tiny x_dbl GEMM, N=96).
// 8 waves/block, each wave 32x32. wm_waves*wn_waves == 8.
// =====================================================================
__global__ __launch_bounds__(256) void wmma_gemm_direct_kernel(
    const float* __restrict__ A, const float* __restrict__ W,
    float* __restrict__ C, int N, int K, int lda, int wm_waves)
{
    const int lane = threadIdx.x & 31;
    const int wave = threadIdx.x >> 5;
    const int wn_waves = 8 / wm_waves;
    const int wm = wave % wm_waves;
    const int wn = wave / wm_waves;

    const int m0 = blockIdx.y * (wm_waves * 32) + wm * 32;
    const int n0 = blockIdx.x * (wn_waves * 32) + wn * 32;
    (void)wn_waves;

    v8f acc00 = {}; v8f acc01 = {}; v8f acc10 = {}; v8f acc11 = {};

    const int l15  = lane & 15;
    const int koff = (lane >> 4) << 1;

    const float* arow0 = A + (size_t)(m0 + l15) * lda + koff;
    const float* arow1 = arow0 + (size_t)16 * lda;
    const float* wrow0 = W + (size_t)(n0 + l15) * K + koff;
    const float* wrow1 = wrow0 + (size_t)16 * K;

#pragma unroll 4
    for (int k = 0; k < K; k += 4) {
        v2f a0 = *(const v2f*)(arow0 + k);
        v2f a1 = *(const v2f*)(arow1 + k);
        v2f b0 = *(const v2f*)(wrow0 + k);
        v2f b1 = *(const v2f*)(wrow1 + k);
        acc00 = wmma4(a0, b0, acc00);
        acc01 = wmma4(a0, b1, acc01);
        acc10 = wmma4(a1, b0, acc10);
        acc11 = wmma4(a1, b1, acc11);
    }

    const int mb  = m0 + ((lane >> 4) << 3);
    const int nc0 = n0 + l15;
    const int nc1 = nc0 + 16;
#pragma unroll
    for (int v = 0; v < 8; ++v) {
        C[(size_t)(mb + v)      * N + nc0] = acc00[v];
        C[(size_t)(mb + v)      * N + nc1] = acc01[v];
        C[(size_t)(mb + 16 + v) * N + nc0] = acc10[v];
        C[(size_t)(mb + 16 + v) * N + nc1] = acc11[v];
    }
}

// =====================================================================
// Causal depthwise conv (taps=4) + bias + SiLU over the xs half of xr.
// =====================================================================
__global__ __launch_bounds__(256) void conv_silu_kernel(
    const float* __restrict__ xr, const float* __restrict__ conv_w,
    const float* __restrict__ conv_b, float* __restrict__ xs_c)
{
    int idx = blockIdx.x * blockDim.x + threadIdx.x;   // over MROWS*INNER
    int d  = idx & (INNER - 1);
    int bt = idx >> 11;            // INNER = 2^11
    int t  = bt & (L_SEQ - 1);
    int b  = bt >> 11;             // L_SEQ = 2^11

    const float w0 = conv_w[d * 4 + 0];
    const float w1 = conv_w[d * 4 + 1];
    const float w2 = conv_w[d * 4 + 2];
    const float w3 = conv_w[d * 4 + 3];

    const size_t base = ((size_t)(b * L_SEQ)) * (2 * INNER) + d;
    float acc = conv_b[d];
    if (t >= 3) acc += w0 * xr[base + (size_t)(t - 3) * (2 * INNER)];
    if (t >= 2) acc += w1 * xr[base + (size_t)(t - 2) * (2 * INNER)];
    if (t >= 1) acc += w2 * xr[base + (size_t)(t - 1) * (2 * INNER)];
    acc += w3 * xr[base + (size_t)t * (2 * INNER)];

    xs_c[(size_t)idx] = silu_f(acc);
}

// =====================================================================
// Selective scan. 16 lanes per (b,d) channel hold the n-state.
// =====================================================================
__global__ __launch_bounds__(256) void scan_kernel(
    const float* __restrict__ xr, const float* __restrict__ xs_c,
    const float* __restrict__ x_dbl, const float* __restrict__ dt2,
    const float* __restrict__ A_log, const float* __restrict__ D_param,
    float* __restrict__ yact)
{
    int g  = blockIdx.x * blockDim.x + threadIdx.x;  // 0 .. B*INNER*NSTATE-1
    int n  = g & (NSTATE - 1);
    int bd = g >> 4;
    int d  = bd & (INNER - 1);
    int b  = bd >> 11;

    const float Acoef = -expf(A_log[d * NSTATE + n]);
    const float Dp    = D_param[d];
    float h = 0.0f;

    for (int t = 0; t < L_SEQ; ++t) {
        const size_t row = (size_t)(b * L_SEQ + t);
        const float dt = dt2 [row * INNER + d];
        const float xv = xs_c[row * INNER + d];
        const float Bt = x_dbl[row * XDBL_N + DTRANK + n];
        const float Ct = x_dbl[row * XDBL_N + DTRANK + NSTATE + n];

        const float dA = expf(dt * Acoef);
        h = dA * h + dt * Bt * xv;
        float p = h * Ct;
        p += __shfl_xor(p, 1, 32);
        p += __shfl_xor(p, 2, 32);
        p += __shfl_xor(p, 4, 32);
        p += __shfl_xor(p, 8, 32);

        if (n == 0) {
            const float res = xr[row * (2 * INNER) + INNER + d];
            yact[row * INNER + d] = (p + xv * Dp) * silu_f(res);
        }
    }
}

// =====================================================================
extern "C" void kernel_launch(void* const* d_in, const int* in_sizes, int n_in,
                              void* d_out, int out_size, void* d_ws, size_t ws_size,
                              hipStream_t stream) {
    (void)in_sizes; (void)n_in; (void)out_size; (void)ws_size;
    const float* x      = (const float*)d_in[0];
    const float* W_in   = (const float*)d_in[1];
    const float* conv_w = (const float*)d_in[2];
    const float* conv_b = (const float*)d_in[3];
    const float* W_x    = (const float*)d_in[4];
    const float* W_dt   = (const float*)d_in[5];
    const float* b_dt   = (const float*)d_in[6];
    const float* A_log  = (const float*)d_in[7];
    const float* D_par  = (const float*)d_in[8];
    const float* W_out  = (const float*)d_in[9];
    float* out = (float*)d_out;

    float* ws    = (float*)d_ws;
    float* xr    = ws;                                   // [4096, 4096]
    float* xs_c  = xr    + (size_t)MROWS * 2 * INNER;    // [4096, 2048]
    float* x_dbl = xs_c  + (size_t)MROWS * INNER;        // [4096, 96]
    float* dt2   = x_dbl + (size_t)MROWS * XDBL_N;       // [4096, 2048]
    float* yact  = dt2   + (size_t)MROWS * INNER;        // [4096, 2048]

    // 1) xr = x @ W_in^T   (M=4096, N=4096, K=1024)
    wmma_gemm_staged_kernel<<<dim3((2 * INNER) / BN, MROWS / BM), 256, 0, stream>>>(
        x, W_in, nullptr, xr, 2 * INNER, DMODEL, DMODEL, 0);

    // 2) depthwise causal conv + SiLU
    conv_silu_kernel<<<(MROWS * INNER) / 256, 256, 0, stream>>>(
        xr, conv_w, conv_b, xs_c);

    // 3) x_dbl = xs_c @ W_x^T  (M=4096, N=96, K=2048) -- tiny, direct path
    wmma_gemm_direct_kernel<<<dim3(XDBL_N / 32, MROWS / 256), 256, 0, stream>>>(
        xs_c, W_x, x_dbl, XDBL_N, INNER, INNER, 8);

    // 4) dt2 = softplus(softplus(x_dbl[:, :64] @ W_dt^T + b_dt))
    //    (M=4096, N=2048, K=64, lda=96)
    wmma_gemm_staged_kernel<<<dim3(INNER / BN, MROWS / BM), 256, 0, stream>>>(
        x_dbl, W_dt, b_dt, dt2, INNER, DTRANK, XDBL_N, 1);

    // 5) selective scan + gating epilogue
    scan_kernel<<<(B_SZ * INNER * NSTATE) / 256, 256, 0, stream>>>(
        xr, xs_c, x_dbl, dt2, A_log, D_par, yact);

    // 6) out = yact @ W_out^T  (M=4096, N=1024, K=2048)
    wmma_gemm_staged_kernel<<<dim3(DMODEL / BN, MROWS / BM), 256, 0, stream>>>(
        yact, W_out, nullptr, out, DMODEL, INNER, INNER, 0);
}